// DiffusionDenoiser_55997783605350
// MI455X (gfx1250) — compile-verified
//
#include <hip/hip_runtime.h>

typedef __attribute__((ext_vector_type(2))) float v2f;
typedef __attribute__((ext_vector_type(8))) float v8f;

#define C 128
#define TM 64          // rows of X per workgroup
#define XS 132         // padded LDS row stride for X tile (floats): +4 mod 64 banks per row
#define WS 136         // padded LDS row stride for W (floats): 2*WS % 64 == 16 -> disjoint half-wave banks

// ---------------------------------------------------------------------------
// t_emb = relu(t * tW1 + tb1) @ tW2 + tb2        (1 block, 128 threads)
// ---------------------------------------------------------------------------
__global__ void time_mlp_kernel(const float* __restrict__ t,
                                const float* __restrict__ tW1, const float* __restrict__ tb1,
                                const float* __restrict__ tW2, const float* __restrict__ tb2,
                                float* __restrict__ t_emb) {
    __shared__ float hid[C];
    int j = threadIdx.x;
    float tv = t[0];
    hid[j] = fmaxf(tv * tW1[j] + tb1[j], 0.0f);
    __syncthreads();
    float acc = tb2[j];
    for (int i = 0; i < C; ++i) acc += hid[i] * tW2[i * C + j];
    t_emb[j] = acc;
}

// ---------------------------------------------------------------------------
// degree (incl. self loop) -> dinv = deg^{-1/2}
// ---------------------------------------------------------------------------
__global__ void deg_init_kernel(float* __restrict__ deg, int n) {
    int i = blockIdx.x * blockDim.x + threadIdx.x;
    if (i < n) deg[i] = 1.0f;                       // self loop
}
__global__ void deg_count_kernel(const int* __restrict__ dst, float* __restrict__ deg, int e) {
    int i = blockIdx.x * blockDim.x + threadIdx.x;
    if (i < e) atomicAdd(&deg[dst[i]], 1.0f);
}
__global__ void dinv_kernel(float* __restrict__ deg, int n) {
    int i = blockIdx.x * blockDim.x + threadIdx.x;
    if (i < n) deg[i] = rsqrtf(deg[i]);
}

// ---------------------------------------------------------------------------
// Y[n_rows x 128] = (X + t_emb?) @ W   using V_WMMA_F32_16X16X4_F32
// 256 threads = 8 wave32; each wave owns one 16-wide column tile and walks
// 4 M-tiles of 16 rows. W staged in LDS (stride 136), X tile in LDS (stride 132).
// ---------------------------------------------------------------------------
__global__ __launch_bounds__(256) void gemm128_wmma_kernel(
        const float* __restrict__ X, const float* __restrict__ W,
        const float* __restrict__ t_emb, float* __restrict__ Y, int n_rows) {
    __shared__ float sW[C * WS];                    // 69,632 B
    __shared__ float sX[TM * XS];                   // 33,792 B

    const int tid  = threadIdx.x;
    const int wave = tid >> 5;
    const int lane = tid & 31;
    const int row0 = blockIdx.x * TM;

    // stage W (128x128) into LDS, padded rows of WS floats
    {
        const float4* W4  = (const float4*)W;
        float4*       sW4 = (float4*)sW;            // WS/4 = 34 float4 per row
        for (int i = tid; i < C * (C / 4); i += 256) {
            int r = i >> 5, c4 = i & 31;
            sW4[r * (WS / 4) + c4] = W4[i];
        }
    }
    // stage X tile (TM x 128) + optional t_emb broadcast, padded rows of XS floats
    {
        float4* sX4 = (float4*)sX;                  // XS/4 = 33 float4 per row
        for (int i = tid; i < TM * (C / 4); i += 256) {
            int r = i >> 5, c4 = i & 31;
            int row = row0 + r;
            float4 v = make_float4(0.f, 0.f, 0.f, 0.f);
            if (row < n_rows) v = ((const float4*)(X + (size_t)row * C))[c4];
            if (t_emb) {
                float4 e = ((const float4*)t_emb)[c4];
                v.x += e.x; v.y += e.y; v.z += e.z; v.w += e.w;
            }
            sX4[r * (XS / 4) + c4] = v;
        }
    }
    __syncthreads();

    const int n0   = wave * 16;                     // column tile
    const int half = lane >> 4;                     // K split within the wave
    const int ml   = lane & 15;

    for (int mt = 0; mt < TM / 16; ++mt) {
        v8f c = {};
        const float* xa = sX + (size_t)(mt * 16 + ml) * XS;   // A row for this lane
        for (int k = 0; k < C; k += 4) {
            v2f a, b;
            // A 16x4 fp32: lanes 0-15 hold {K,K+1}, lanes 16-31 hold {K+2,K+3}
            a[0] = xa[k + 2 * half];
            a[1] = xa[k + 2 * half + 1];
            // B 4x16 fp32: VGPR0 rows K / K+2, VGPR1 rows K+1 / K+3, cols across lanes
            b[0] = sW[(k + 2 * half)     * WS + n0 + ml];
            b[1] = sW[(k + 2 * half + 1) * WS + n0 + ml];
            c = __builtin_amdgcn_wmma_f32_16x16x4_f32(
                    false, a, false, b, (short)0, c, false, false);
        }
        // C/D 16x16 fp32: VGPR r holds rows r (lanes 0-15) and r+8 (lanes 16-31)
        int row_lo = row0 + mt * 16 + half * 8;
        int col    = n0 + ml;
        for (int r = 0; r < 8; ++r) {
            int row = row_lo + r;
            if (row < n_rows) Y[(size_t)row * C + col] = c[r];
        }
    }
}

// ---------------------------------------------------------------------------
// AGG = XW * dinv^2 (self loop) + bias   -- also initializes the output buffer
// ---------------------------------------------------------------------------
__global__ void self_bias_kernel(const float* __restrict__ XW, const float* __restrict__ dinv,
                                 const float* __restrict__ bias, float* __restrict__ AGG, int n) {
    size_t i = (size_t)blockIdx.x * blockDim.x + threadIdx.x;
    if (i < (size_t)n * C) {
        int row = (int)(i >> 7);
        int ch  = (int)(i & (C - 1));
        float di = dinv[row];
        AGG[i] = XW[i] * di * di + bias[ch];
    }
}

// ---------------------------------------------------------------------------
// scatter-add: one wave per edge, 4 channels per lane (float4 gather, 4 atomics)
// ---------------------------------------------------------------------------
__global__ void scatter_kernel(const float* __restrict__ XW, const float* __restrict__ dinv,
                               const int* __restrict__ src, const int* __restrict__ dst,
                               float* __restrict__ AGG, int e) {
    int gid  = blockIdx.x * blockDim.x + threadIdx.x;
    int edge = gid >> 5;
    int lane = gid & 31;
    if (edge >= e) return;
    int s = src[edge], d = dst[edge];
    float coef = dinv[s] * dinv[d];
    float* outp = AGG + (size_t)d * C + lane * 4;
    __builtin_prefetch(outp, 1, 3);                 // global_prefetch_b8 on dst row
    float4 v = ((const float4*)(XW + (size_t)s * C))[lane];
    atomicAdd(outp + 0, v.x * coef);
    atomicAdd(outp + 1, v.y * coef);
    atomicAdd(outp + 2, v.z * coef);
    atomicAdd(outp + 3, v.w * coef);
}

// ---------------------------------------------------------------------------
// in-place LayerNorm + ReLU, one wave32 per row, 4 channels/lane
// ---------------------------------------------------------------------------
__global__ void ln_relu_kernel(float* __restrict__ H, const float* __restrict__ ln_w,
                               const float* __restrict__ ln_b, int n) {
    int gid  = blockIdx.x * blockDim.x + threadIdx.x;
    int row  = gid >> 5;
    int lane = gid & 31;
    if (row >= n) return;
    float4 v = ((float4*)(H + (size_t)row * C))[lane];
    float s = v.x + v.y + v.z + v.w;
    for (int m = 16; m; m >>= 1) s += __shfl_xor(s, m, 32);
    float mu = s * (1.0f / C);
    float dx = v.x - mu, dy = v.y - mu, dz = v.z - mu, dw = v.w - mu;
    float q = dx * dx + dy * dy + dz * dz + dw * dw;
    for (int m = 16; m; m >>= 1) q += __shfl_xor(q, m, 32);
    float rinv = rsqrtf(q * (1.0f / C) + 1e-5f);
    float4 w4 = ((const float4*)ln_w)[lane];
    float4 b4 = ((const float4*)ln_b)[lane];
    float4 o;
    o.x = fmaxf(dx * rinv * w4.x + b4.x, 0.0f);
    o.y = fmaxf(dy * rinv * w4.y + b4.y, 0.0f);
    o.z = fmaxf(dz * rinv * w4.z + b4.z, 0.0f);
    o.w = fmaxf(dw * rinv * w4.w + b4.w, 0.0f);
    ((float4*)(H + (size_t)row * C))[lane] = o;
}

// ---------------------------------------------------------------------------
extern "C" void kernel_launch(void* const* d_in, const int* in_sizes, int n_in,
                              void* d_out, int out_size, void* d_ws, size_t ws_size,
                              hipStream_t stream) {
    const float* h_noisy = (const float*)d_in[0];
    const int*   eidx    = (const int*)d_in[1];
    const float* t       = (const float*)d_in[2];
    const float* tW1     = (const float*)d_in[3];
    const float* tb1     = (const float*)d_in[4];
    const float* tW2     = (const float*)d_in[5];
    const float* tb2     = (const float*)d_in[6];
    const float* W1      = (const float*)d_in[7];
    const float* b1      = (const float*)d_in[8];
    const float* W2      = (const float*)d_in[9];
    const float* b2      = (const float*)d_in[10];
    const float* ln_w    = (const float*)d_in[11];
    const float* ln_b    = (const float*)d_in[12];

    const int N = in_sizes[0] / C;
    const int E = in_sizes[1] / 2;
    const int* src = eidx;
    const int* dst = eidx + E;

    float* ws    = (float*)d_ws;
    float* t_emb = ws;                              // 128
    float* dinv  = ws + C;                          // N
    size_t o     = (size_t)C + (size_t)N;
    o = (o + 3) & ~(size_t)3;                       // float4 align
    float* bufA  = ws + o;                          // N*128  (XW)
    float* bufB  = bufA + (size_t)N * C;            // N*128  (AGG / h1)
    float* out   = (float*)d_out;

    const int BT = 256;
    dim3 blk(BT);

    // time embedding
    hipLaunchKernelGGL(time_mlp_kernel, dim3(1), dim3(C), 0, stream,
                       t, tW1, tb1, tW2, tb2, t_emb);
    // degrees -> dinv
    hipLaunchKernelGGL(deg_init_kernel, dim3((N + BT - 1) / BT), blk, 0, stream, dinv, N);
    hipLaunchKernelGGL(deg_count_kernel, dim3((E + BT - 1) / BT), blk, 0, stream, dst, dinv, E);
    hipLaunchKernelGGL(dinv_kernel, dim3((N + BT - 1) / BT), blk, 0, stream, dinv, N);

    const int gemm_blocks = (N + TM - 1) / TM;
    const int nc_blocks   = (int)(((size_t)N * C + BT - 1) / BT);
    const int ew_blocks   = (int)(((size_t)E * 32 + BT - 1) / BT);
    const int row_blocks  = (int)(((size_t)N * 32 + BT - 1) / BT);

    // ---- layer 1: h = relu(LN(GCN(h_noisy + t_emb))) ----
    hipLaunchKernelGGL(gemm128_wmma_kernel, dim3(gemm_blocks), blk, 0, stream,
                       h_noisy, W1, t_emb, bufA, N);
    hipLaunchKernelGGL(self_bias_kernel, dim3(nc_blocks), blk, 0, stream,
                       bufA, dinv, b1, bufB, N);
    hipLaunchKernelGGL(scatter_kernel, dim3(ew_blocks), blk, 0, stream,
                       bufA, dinv, src, dst, bufB, E);
    hipLaunchKernelGGL(ln_relu_kernel, dim3(row_blocks), blk, 0, stream,
                       bufB, ln_w, ln_b, N);

    // ---- layer 2: out = GCN(h) ----
    hipLaunchKernelGGL(gemm128_wmma_kernel, dim3(gemm_blocks), blk, 0, stream,
                       bufB, W2, (const float*)nullptr, bufA, N);
    hipLaunchKernelGGL(self_bias_kernel, dim3(nc_blocks), blk, 0, stream,
                       bufA, dinv, b2, out, N);
    hipLaunchKernelGGL(scatter_kernel, dim3(ew_blocks), blk, 0, stream,
                       bufA, dinv, src, dst, out, E);
}